// Attention_69638599737644
// MI455X (gfx1250) — compile-verified
//
#include <hip/hip_runtime.h>

// ---------------------------------------------------------------------------
// Cosine linear attention for MI455X (gfx1250, wave32, WMMA bf16 16x16x32).
// N=16, C=1024, d=1024, H=16, hd=64.
// - Tiles staged in LDS via GLOBAL_LOAD_ASYNC_TO_LDS_B128 (ASYNCcnt): no VGPR
//   transit, no ds_store, staging latency hidden behind the WMMAs.
// - Fragments loaded as two aligned uint4 (guaranteed 2x ds_load_b128).
// - WMMA issued as inline asm with tied D=C operand so every accumulator
//   stays pinned in one 8-VGPR tuple (kills the v_mov_b64 regalloc shuffling).
// - 256x64 macro-tiles, 32x64 per wave, ping-pong LDS, 1 barrier per k-step.
// - Transposes folded into epilogue stores / natural W[n][k] layout.
// ---------------------------------------------------------------------------

typedef __attribute__((ext_vector_type(16))) __bf16 v16bf;
typedef __attribute__((ext_vector_type(8)))  float  v8f;

#define NHEADS 16
#define HD     64
#define CDIM   1024
#define DDIM   1024
#define NBATCH 16
#define MTOT   (NBATCH * CDIM)   // 16384

// LDS byte offset of a __shared__ pointer (generic LDS address low 32 bits).
__device__ __forceinline__ unsigned lds_off(const void* p) {
  return (unsigned)(unsigned long long)p;
}

// Async copy: 16 bytes / lane, memory -> LDS, tracked by ASYNCcnt.
__device__ __forceinline__ void async_b128(unsigned lds, const void* gaddr) {
  asm volatile("global_load_async_to_lds_b128 %0, %1, off"
               :: "v"(lds), "v"(gaddr) : "memory");
}
__device__ __forceinline__ void async_b128x4(unsigned lds, const void* gaddr) {
  asm volatile("global_load_async_to_lds_b128 %0, %1, off\n\t"
               "global_load_async_to_lds_b128 %0, %1, off offset:16\n\t"
               "global_load_async_to_lds_b128 %0, %1, off offset:32\n\t"
               "global_load_async_to_lds_b128 %0, %1, off offset:48"
               :: "v"(lds), "v"(gaddr) : "memory");
}
__device__ __forceinline__ void wait_async0() {
  asm volatile("s_wait_asynccnt 0x0" ::: "memory");
}

// Fragment load (ISA 7.12.2 16-bit 16x32 layout): two contiguous 8-element K
// runs per lane -> exactly 2x ds_load_b128.
__device__ __forceinline__ v16bf frag16(const __bf16* lds, int row0, int stride, int lane) {
  const int m = lane & 15, half = lane >> 4;
  const __bf16* p = lds + (row0 + m) * stride + half * 8;
  union { v16bf v; uint4 q[2]; } u;
  u.q[0] = *(const uint4*)(p);
  u.q[1] = *(const uint4*)(p + 16);
  return u.v;
}

// WMMA with tied D=C: accumulator pinned to one even-aligned 8-VGPR tuple.
// D->C accumulation chains need no NOPs (ISA 7.12.1 lists only D->A/B).
__device__ __forceinline__ void wmma_acc(v8f& c, const v16bf& a, const v16bf& b) {
  asm("v_wmma_f32_16x16x32_bf16 %0, %1, %2, %0"
      : "+v"(c)
      : "v"(a), "v"(b));
}
// WMMA -> VALU read-of-D hazard cover (4 co-exec NOPs for bf16 WMMA).
__device__ __forceinline__ void wmma_fence() {
  asm volatile("s_nop 4" :::);
}

// ---------------------------------------------------------------------------
// f32 -> bf16 conversion (x and all four weights; layouts preserved)
// ---------------------------------------------------------------------------
__global__ void cvt_bf16(const float* __restrict__ x, __bf16* __restrict__ y, int n) {
  int i = (blockIdx.x * blockDim.x + threadIdx.x) * 4;
  if (i < n) {
    float4 v = *(const float4*)(x + i);
    y[i + 0] = (__bf16)v.x;
    y[i + 1] = (__bf16)v.y;
    y[i + 2] = (__bf16)v.z;
    y[i + 3] = (__bf16)v.w;
  }
}

// ---------------------------------------------------------------------------
// QKV projection GEMM, 256x64 macro-tile, async double-buffered, fused
// epilogues.  grid: (MTOT/256, NHEADS, 3)  block: 256 (8 waves, each 32x64)
// z=0 -> q (L2 norm)   -> qn  [N,H,C,hd]   (A-natural for attn kernel)
// z=1 -> k (L2 norm)   -> knT [N,H,hd,C]   (K-contiguous for kv kernel)
// z=2 -> v (sig scale) -> vsT [N,H,hd,C]   (K-contiguous for kv kernel)
// ---------------------------------------------------------------------------
#define QKV_NK (DDIM / 32)

__global__ __launch_bounds__(256) void qkv_kernel(
    const __bf16* __restrict__ A,    // xb  [16384][1024]
    const __bf16* __restrict__ W3,   // 3 x [1024][1024]  (Wq, Wk, Wv, n-major)
    __bf16* __restrict__ qn, __bf16* __restrict__ knT, __bf16* __restrict__ vsT,
    const float* __restrict__ norm_const)
{
  __shared__ __align__(16) __bf16 sA[2][256 * 32];   // 2 x 16 KB
  __shared__ __align__(16) __bf16 sB[2][64 * 32];    // 2 x  4 KB
  const int m0   = blockIdx.x * 256;
  const int head = blockIdx.y;
  const int n0   = head * HD;
  const int z    = blockIdx.z;
  const __bf16* B = W3 + (size_t)z * DDIM * DDIM;
  const int t = threadIdx.x, lane = t & 31, w = t >> 5;

  v8f acc[2][4];
  const v8f vz = {0.f, 0.f, 0.f, 0.f, 0.f, 0.f, 0.f, 0.f};
#pragma unroll
  for (int s = 0; s < 2; ++s)
#pragma unroll
    for (int j = 0; j < 4; ++j) acc[s][j] = vz;

  const int br = t >> 2, bc = (t & 3) * 8;           // B stage: 64 rows x 32 K
  const __bf16* aRow = A + (size_t)(m0 + t) * DDIM;  // A stage: 1 row / thread
  const __bf16* bRow = B + (size_t)(n0 + br) * DDIM + bc;
  const unsigned la0 = lds_off(&sA[0][t * 32]);
  const unsigned lb0 = lds_off(&sB[0][br * 32 + bc]);

  async_b128x4(la0, aRow);
  async_b128(lb0, bRow);
  wait_async0();
  __syncthreads();

  for (int kt = 0; kt < QKV_NK; ++kt) {
    const int cur = kt & 1;
    if (kt + 1 < QKV_NK) {                 // stage next tile into free buffer
      async_b128x4(la0 + (1 - cur) * (256 * 32 * 2), aRow + (kt + 1) * 32);
      async_b128(lb0 + (1 - cur) * (64 * 32 * 2), bRow + (kt + 1) * 32);
    }
    const __bf16* pa = sA[cur];
    const __bf16* pb = sB[cur];
    const v16bf fa0 = frag16(pa, 32 * w, 32, lane);
    const v16bf fa1 = frag16(pa, 32 * w + 16, 32, lane);
#pragma unroll
    for (int j = 0; j < 4; ++j) {
      const v16bf fb = frag16(pb, 16 * j, 32, lane);
      wmma_acc(acc[0][j], fa0, fb);
      wmma_acc(acc[1][j], fa1, fb);
    }
    wait_async0();                         // after MMAs: latency already hidden
    __syncthreads();
  }
  wmma_fence();

  const int half = lane >> 4, nloc = lane & 15;
  if (z < 2) {
#pragma unroll
    for (int s = 0; s < 2; ++s)
#pragma unroll
      for (int r = 0; r < 8; ++r) {
        float sq = 0.f;                    // full head dim is wave-resident
#pragma unroll
        for (int j = 0; j < 4; ++j) sq += acc[s][j][r] * acc[s][j][r];
        sq += __shfl_xor(sq, 1, 32);
        sq += __shfl_xor(sq, 2, 32);
        sq += __shfl_xor(sq, 4, 32);
        sq += __shfl_xor(sq, 8, 32);       // reduce in 16-lane half-group
        const float scale = 1.0f / fmaxf(sqrtf(sq), 1e-12f);
        const int grow = m0 + 32 * w + 16 * s + r + 8 * half;
        const int nb = grow >> 10, c = grow & 1023;
        const size_t bh = (size_t)(nb * NHEADS + head);
        if (z == 0) {
          __bf16* dst = qn + (bh * CDIM + c) * HD;
#pragma unroll
          for (int j = 0; j < 4; ++j) dst[16 * j + nloc] = (__bf16)(acc[s][j][r] * scale);
        } else {
#pragma unroll
          for (int j = 0; j < 4; ++j)      // transposed store: [bh][d][c]
            knT[(bh * HD + 16 * j + nloc) * CDIM + c] = (__bf16)(acc[s][j][r] * scale);
        }
      }
  } else {
    const float nc = norm_const[head];
    const float sg = 1.f / (1.f + __expf(-nc));
    const float scale = __expf(-sg * __logf((float)CDIM));  // C^-sigmoid(nc)
#pragma unroll
    for (int s = 0; s < 2; ++s)
#pragma unroll
      for (int r = 0; r < 8; ++r) {
        const int grow = m0 + 32 * w + 16 * s + r + 8 * half;
        const int nb = grow >> 10, c = grow & 1023;
        const size_t bh = (size_t)(nb * NHEADS + head);
#pragma unroll
        for (int j = 0; j < 4; ++j)        // transposed store: [bh][e][c]
          vsT[(bh * HD + 16 * j + nloc) * CDIM + c] = (__bf16)(acc[s][j][r] * scale);
      }
  }
}

// ---------------------------------------------------------------------------
// kv = kᵀ v  per (n,h): [64 x 1024] x [1024 x 64] -> kvbT [e][d]
// Both operands K(=c)-contiguous; async double-buffered.
// grid: N*H = 256 blocks, block 128 (4 waves, each a 16-row strip of d)
// ---------------------------------------------------------------------------
__global__ __launch_bounds__(128) void kv_kernel(
    const __bf16* __restrict__ knT, const __bf16* __restrict__ vsT,
    __bf16* __restrict__ kvbT)
{
  __shared__ __align__(16) __bf16 sK[2][64 * 32];
  __shared__ __align__(16) __bf16 sV[2][64 * 32];
  const int bh = blockIdx.x;
  const int t = threadIdx.x, lane = t & 31, w = t >> 5;
  const int r = t >> 1, h = (t & 1) * 16;            // 64 rows x 32 c stage
  const __bf16* kRow = knT + (size_t)(bh * HD + r) * CDIM + h;
  const __bf16* vRow = vsT + (size_t)(bh * HD + r) * CDIM + h;
  const unsigned lk0 = lds_off(&sK[0][r * 32 + h]);
  const unsigned lv0 = lds_off(&sV[0][r * 32 + h]);

  v8f acc[4];
  const v8f vz = {0.f, 0.f, 0.f, 0.f, 0.f, 0.f, 0.f, 0.f};
#pragma unroll
  for (int j = 0; j < 4; ++j) acc[j] = vz;

  async_b128(lk0, kRow);  async_b128(lk0 + 16, kRow + 8);
  async_b128(lv0, vRow);  async_b128(lv0 + 16, vRow + 8);
  wait_async0();
  __syncthreads();

  for (int ct = 0; ct < CDIM / 32; ++ct) {
    const int cur = ct & 1;
    if (ct + 1 < CDIM / 32) {
      const unsigned bo = (1 - cur) * (64 * 32 * 2);
      async_b128(lk0 + bo, kRow + (ct + 1) * 32);
      async_b128(lk0 + bo + 16, kRow + (ct + 1) * 32 + 8);
      async_b128(lv0 + bo, vRow + (ct + 1) * 32);
      async_b128(lv0 + bo + 16, vRow + (ct + 1) * 32 + 8);
    }
    const v16bf fa = frag16(sK[cur], 16 * w, 32, lane);    // A rows = d
#pragma unroll
    for (int j = 0; j < 4; ++j)
      wmma_acc(acc[j], fa, frag16(sV[cur], 16 * j, 32, lane));
    wait_async0();
    __syncthreads();
  }
  wmma_fence();

  const int half = lane >> 4, nloc = lane & 15;
#pragma unroll
  for (int rr = 0; rr < 8; ++rr) {
    const int d = 16 * w + rr + 8 * half;
#pragma unroll
    for (int j = 0; j < 4; ++j) {          // transposed store: [bh][e][d]
      const int e = 16 * j + nloc;
      kvbT[((size_t)bh * HD + e) * HD + d] = (__bf16)acc[j][rr];
    }
  }
}

// ---------------------------------------------------------------------------
// attn = q @ kv per (n,h): [1024 x 64] x [64 x 64]; K=64 fits in one stage
// (single barrier, 16 WMMA per wave). Heads re-concatenated so the result is
// a [16384 x 1024] bf16 A-operand for the output projection.
// grid: (C/256, N*H), block 256 (8 waves, each 32x64)
// ---------------------------------------------------------------------------
__global__ __launch_bounds__(256) void attn_kernel(
    const __bf16* __restrict__ qn, const __bf16* __restrict__ kvbT,
    __bf16* __restrict__ attnb)
{
  __shared__ __align__(16) __bf16 sA[256 * 64];   // 32 KB: full K range
  __shared__ __align__(16) __bf16 sB[64 * 64];    //  8 KB
  const int m0 = blockIdx.x * 256;                // token rows within C
  const int bh = blockIdx.y;
  const int nb = bh >> 4, head = bh & 15;
  const __bf16* Ab = qn + (size_t)bh * CDIM * HD;
  const __bf16* Bb = kvbT + (size_t)bh * HD * HD;
  const int t = threadIdx.x, lane = t & 31, w = t >> 5;

  v8f acc[2][4];
  const v8f vz = {0.f, 0.f, 0.f, 0.f, 0.f, 0.f, 0.f, 0.f};
#pragma unroll
  for (int s = 0; s < 2; ++s)
#pragma unroll
    for (int j = 0; j < 4; ++j) acc[s][j] = vz;

  // A: first 32 bf16 of one row (4x b128) / thread; B: 16 bf16 (2x b128) / thread
  async_b128x4(lds_off(&sA[t * 64]), Ab + (size_t)(m0 + t) * HD);
  {
    const int r = t >> 2, c = (t & 3) * 16;
    const unsigned lb = lds_off(&sB[r * 64 + c]);
    const __bf16* src = Bb + (size_t)r * HD + c;
    async_b128(lb, src);
    async_b128(lb + 16, src + 8);
  }
  wait_async0();
  __syncthreads();

#pragma unroll
  for (int kt = 0; kt < 2; ++kt) {        // K chunks at base offsets 0, 32
    const __bf16* pa = sA + kt * 32;
    const __bf16* pb = sB + kt * 32;
    const v16bf fa0 = frag16(pa, 32 * w, 64, lane);
    const v16bf fa1 = frag16(pa, 32 * w + 16, 64, lane);
#pragma unroll
    for (int j = 0; j < 4; ++j) {
      const v16bf fb = frag16(pb, 16 * j, 64, lane);
      wmma_acc(acc[0][j], fa0, fb);
      wmma_acc(acc[1][j], fa1, fb);
    }
  }
  wmma_fence();

  const int half = lane >> 4, nloc = lane & 15;
#pragma unroll
  for (int s = 0; s < 2; ++s)
#pragma unroll
    for (int r = 0; r < 8; ++r) {
      const int grow = m0 + 32 * w + 16 * s + r + 8 * half;   // c index
      __bf16* dst = attnb + ((size_t)(nb * CDIM + grow)) * DDIM + head * HD;
#pragma unroll
      for (int j = 0; j < 4; ++j) dst[16 * j + nloc] = (__bf16)acc[s][j][r];
    }
}

// ---------------------------------------------------------------------------
// out = attn @ Woᵀ : [16384 x 1024] x [1024 x 1024] -> f32 d_out
// B = Wo[n][k] natural layout; 256x64 macro-tile, async double-buffered.
// grid: (MTOT/256, DDIM/64), block 256
// ---------------------------------------------------------------------------
__global__ __launch_bounds__(256) void out_kernel(
    const __bf16* __restrict__ A, const __bf16* __restrict__ Wo,
    float* __restrict__ out)
{
  __shared__ __align__(16) __bf16 sA[2][256 * 32];
  __shared__ __align__(16) __bf16 sB[2][64 * 32];
  const int m0 = blockIdx.x * 256;
  const int n0 = blockIdx.y * 64;
  const int t = threadIdx.x, lane = t & 31, w = t >> 5;

  v8f acc[2][4];
  const v8f vz = {0.f, 0.f, 0.f, 0.f, 0.f, 0.f, 0.f, 0.f};
#pragma unroll
  for (int s = 0; s < 2; ++s)
#pragma unroll
    for (int j = 0; j < 4; ++j) acc[s][j] = vz;

  const int br = t >> 2, bc = (t & 3) * 8;
  const __bf16* aRow = A + (size_t)(m0 + t) * DDIM;
  const __bf16* bRow = Wo + (size_t)(n0 + br) * DDIM + bc;
  const unsigned la0 = lds_off(&sA[0][t * 32]);
  const unsigned lb0 = lds_off(&sB[0][br * 32 + bc]);

  async_b128x4(la0, aRow);
  async_b128(lb0, bRow);
  wait_async0();
  __syncthreads();

  for (int kt = 0; kt < QKV_NK; ++kt) {
    const int cur = kt & 1;
    if (kt + 1 < QKV_NK) {
      async_b128x4(la0 + (1 - cur) * (256 * 32 * 2), aRow + (kt + 1) * 32);
      async_b128(lb0 + (1 - cur) * (64 * 32 * 2), bRow + (kt + 1) * 32);
    }
    const __bf16* pa = sA[cur];
    const __bf16* pb = sB[cur];
    const v16bf fa0 = frag16(pa, 32 * w, 32, lane);
    const v16bf fa1 = frag16(pa, 32 * w + 16, 32, lane);
#pragma unroll
    for (int j = 0; j < 4; ++j) {
      const v16bf fb = frag16(pb, 16 * j, 32, lane);
      wmma_acc(acc[0][j], fa0, fb);
      wmma_acc(acc[1][j], fa1, fb);
    }
    wait_async0();
    __syncthreads();
  }
  wmma_fence();

  const int half = lane >> 4, nloc = lane & 15;
#pragma unroll
  for (int s = 0; s < 2; ++s)
#pragma unroll
    for (int r = 0; r < 8; ++r) {
      const int grow = m0 + 32 * w + 16 * s + r + 8 * half;
      float* dst = out + (size_t)grow * DDIM + n0;
#pragma unroll
      for (int j = 0; j < 4; ++j) dst[16 * j + nloc] = acc[s][j][r];
    }
}

// ---------------------------------------------------------------------------
// Launch
// ---------------------------------------------------------------------------
extern "C" void kernel_launch(void* const* d_in, const int* in_sizes, int n_in,
                              void* d_out, int out_size, void* d_ws, size_t ws_size,
                              hipStream_t stream) {
  (void)in_sizes; (void)n_in; (void)out_size; (void)ws_size;
  const float* x  = (const float*)d_in[0];
  const float* Wq = (const float*)d_in[1];
  const float* Wk = (const float*)d_in[2];
  const float* Wv = (const float*)d_in[3];
  const float* Wo = (const float*)d_in[4];
  const float* nc = (const float*)d_in[5];
  float* out = (float*)d_out;

  // Workspace layout (bytes)
  char* ws = (char*)d_ws;
  const size_t XB_BYTES = (size_t)MTOT * DDIM * 2;                 // 32 MB
  const size_t W_BYTES  = (size_t)DDIM * DDIM * 2;                 // 2 MB each
  const size_t QN_BYTES = (size_t)NBATCH * NHEADS * CDIM * HD * 2; // 32 MB each
  __bf16* xb    = (__bf16*)(ws);
  __bf16* W3    = (__bf16*)(ws + XB_BYTES);                        // Wq,Wk,Wv
  __bf16* Wob   = (__bf16*)(ws + XB_BYTES + 3 * W_BYTES);
  __bf16* qn    = (__bf16*)(ws + XB_BYTES + 4 * W_BYTES);
  __bf16* knT   = (__bf16*)(ws + XB_BYTES + 4 * W_BYTES + QN_BYTES);
  __bf16* vsT   = (__bf16*)(ws + XB_BYTES + 4 * W_BYTES + 2 * QN_BYTES);
  __bf16* kvbT  = (__bf16*)(ws + XB_BYTES + 4 * W_BYTES + 3 * QN_BYTES);
  __bf16* attnb = xb;   // x is dead after qkv_kernel; reuse its slot

  // 1) bf16 conversions (layouts preserved; W[n][k] is already B-operand form)
  {
    const int nx = MTOT * DDIM;
    cvt_bf16<<<dim3(nx / 4 / 256), dim3(256), 0, stream>>>(x, xb, nx);
    const int nw = DDIM * DDIM;
    const dim3 gw(nw / 4 / 256), bw(256);
    cvt_bf16<<<gw, bw, 0, stream>>>(Wq, W3, nw);
    cvt_bf16<<<gw, bw, 0, stream>>>(Wk, W3 + (size_t)DDIM * DDIM, nw);
    cvt_bf16<<<gw, bw, 0, stream>>>(Wv, W3 + 2 * (size_t)DDIM * DDIM, nw);
    cvt_bf16<<<gw, bw, 0, stream>>>(Wo, Wob, nw);
  }
  // 2) q/k/v projections + fused normalize / scale (k,v stored K-contiguous)
  qkv_kernel<<<dim3(MTOT / 256, NHEADS, 3), dim3(256), 0, stream>>>(
      xb, W3, qn, knT, vsT, nc);
  // 3) kv = kᵀ v  (output stored transposed for attn's B operand)
  kv_kernel<<<dim3(NBATCH * NHEADS), dim3(128), 0, stream>>>(knT, vsT, kvbT);
  // 4) attn = q @ kv  (head-concat layout, A-operand form for out)
  attn_kernel<<<dim3(CDIM / 256, NBATCH * NHEADS), dim3(256), 0, stream>>>(
      qn, kvbT, attnb);
  // 5) out = attn @ Wo  (f32)
  out_kernel<<<dim3(MTOT / 256, DDIM / 64), dim3(256), 0, stream>>>(
      attnb, Wob, out);
}